// SpatioPositionalModule_13494787244854
// MI455X (gfx1250) — compile-verified
//
#include <hip/hip_runtime.h>
#include <math.h>

typedef __attribute__((ext_vector_type(2))) float v2f;
typedef __attribute__((ext_vector_type(8))) float v8f;
typedef __attribute__((ext_vector_type(4))) int v4i;

// pointer-to-AS1(v4i) / pointer-to-AS3(v4i), matching the async builtin params
typedef __attribute__((address_space(1))) v4i* gbl_v4i_p;
typedef __attribute__((address_space(3))) v4i* lds_v4i_p;

#define S_DIM 4096
#define FEAT 1024
#define ATTN 1024
#define SIGMA_X 4.0f
#define SIGMA_Y 1.0f
#define EPS_VALID 1e-8f
#define EPS_DIS 1e-4f

#define KC 16        // K-chunk for targets GEMM staging
#define AS (KC + 4)  // padded LDS row stride for A tile (bank-conflict break)
#define KJ 32        // J-chunk for output GEMM staging

#if __has_builtin(__builtin_amdgcn_global_load_async_to_lds_b128)
#define HAVE_ASYNC_LDS 1
#endif

// ---------------------------------------------------------------------------
// CDNA5 helpers
// ---------------------------------------------------------------------------
__device__ __forceinline__ v8f wmma4(v2f a, v2f b, v8f c) {
  return __builtin_amdgcn_wmma_f32_16x16x4_f32(
      /*neg_a=*/false, a, /*neg_b=*/false, b,
      /*c_mod=*/(short)0, c, /*reuse_a=*/false, /*reuse_b=*/false);
}

// one float4 per lane, global -> LDS through the async engine (ASYNCcnt)
__device__ __forceinline__ void lds_copy_f4(const float* g, float* l) {
#if defined(HAVE_ASYNC_LDS)
  __builtin_amdgcn_global_load_async_to_lds_b128(
      (gbl_v4i_p)(void*)g, (lds_v4i_p)(void*)l,
      /*imm offset=*/0, /*cpol=*/0);
#else
  *(float4*)l = *(const float4*)g;
#endif
}

__device__ __forceinline__ void wait_async() {
#if defined(HAVE_ASYNC_LDS)
#if __has_builtin(__builtin_amdgcn_s_wait_asynccnt)
  __builtin_amdgcn_s_wait_asynccnt(0);
#else
  asm volatile("s_wait_asynccnt 0x0" ::: "memory");
#endif
#endif
}

// ---------------------------------------------------------------------------
// Kernel A: highlight head. One workgroup (256 threads) per row.
// ---------------------------------------------------------------------------
__global__ void highlight_kernel(const float* __restrict__ X,
                                 const float* __restrict__ xywh,
                                 const float* __restrict__ Wh,
                                 const float* __restrict__ bh,
                                 const int* __restrict__ num_person,
                                 float* __restrict__ out_hp,
                                 float* __restrict__ hx,
                                 float* __restrict__ hy,
                                 float* __restrict__ validf) {
  __shared__ float s0[256];
  __shared__ float s1[256];
  const int i = blockIdx.x;
  const int t = threadIdx.x;

  const float4 xv = ((const float4*)(X + (size_t)i * FEAT))[t];
  const float2* wh = (const float2*)Wh;
  const float2 w0 = wh[t * 4 + 0];
  const float2 w1 = wh[t * 4 + 1];
  const float2 w2 = wh[t * 4 + 2];
  const float2 w3 = wh[t * 4 + 3];

  s0[t] = xv.x * w0.x + xv.y * w1.x + xv.z * w2.x + xv.w * w3.x;
  s1[t] = xv.x * w0.y + xv.y * w1.y + xv.z * w2.y + xv.w * w3.y;
  __syncthreads();
  for (int off = 128; off > 0; off >>= 1) {
    if (t < off) {
      s0[t] += s0[t + off];
      s1[t] += s1[t + off];
    }
    __syncthreads();
  }
  if (t == 0) {
    const float h0 = tanhf(s0[0] + bh[0]);
    const float h1 = tanhf(s1[0] + bh[1]);
    const float bx = xywh[i * 4 + 0];
    const float by = xywh[i * 4 + 1];
    const float bw = xywh[i * 4 + 2];
    const float bhh = xywh[i * 4 + 3];
    out_hp[i * 2 + 0] = h0;
    out_hp[i * 2 + 1] = h1;
    hx[i] = h0 * bw * SIGMA_X + bx;
    hy[i] = h1 * bhh * SIGMA_Y + by;
    const float sum = bx + by + bw + bhh;
    validf[i] = (sum >= EPS_VALID && i < num_person[0]) ? 1.0f : 0.0f;
  }
}

// ---------------------------------------------------------------------------
// Kernel B: targets = relu(X @ Wg + bg), f32 WMMA 16x16x4, async-LDS staged.
// Block = (32,4): 4 waves share one 16-row A tile; each wave owns 64 N-cols.
// Double-buffered chunks of K=KC staged via GLOBAL_LOAD_ASYNC_TO_LDS_B128.
// ---------------------------------------------------------------------------
__global__ void targets_gemm_kernel(const float* __restrict__ X,
                                    const float* __restrict__ Wg,
                                    const float* __restrict__ bg,
                                    float* __restrict__ T) {
  __shared__ float Ab[2][16 * AS];    // 16 x KC (padded rows)
  __shared__ float Bb[2][KC * 256];   // KC x 256

  const int lane = threadIdx.x;
  const int wave = threadIdx.y;
  const int tid = wave * 32 + lane;   // 0..127
  const int mBase = blockIdx.y * 16;
  const int nBase0 = blockIdx.x * 256;
  const int ml = lane & 15;
  const int kh = (lane >> 4) * 2;     // 0 or 2

  v8f acc0 = {}, acc1 = {}, acc2 = {}, acc3 = {};

  auto stage = [&](int b, int kc) {
    // A tile: 16*KC/4 float4 (= 64 for KC=16) -> lanes of waves 0..1
    if (tid < (16 * KC / 4)) {
      const int r = tid / (KC / 4);
      const int c4 = (tid % (KC / 4)) * 4;
      lds_copy_f4(X + (size_t)(mBase + r) * FEAT + kc + c4, &Ab[b][r * AS + c4]);
    }
    // B tile: KC*256/4 float4, spread over 128 threads
#pragma unroll
    for (int t = 0; t < (KC * 256 / 4) / 128; ++t) {
      const int q = t * 128 + tid;
      const int kr = q / 64;
      const int n4 = (q % 64) * 4;
      lds_copy_f4(Wg + (size_t)(kc + kr) * ATTN + nBase0 + n4,
                  &Bb[b][kr * 256 + n4]);
    }
  };

  auto compute = [&](int b) {
    const float* Arow = &Ab[b][0] + ml * AS;
    const float* Bbase = &Bb[b][0] + wave * 64 + ml;
#pragma unroll
    for (int k = 0; k < KC; k += 4) {
      v2f a;
      a.x = Arow[k + kh];
      a.y = Arow[k + kh + 1];
      const float* p0 = Bbase + (k + kh) * 256;
      const float* p1 = p0 + 256;
      v2f bb;
      bb.x = p0[0];  bb.y = p1[0];  acc0 = wmma4(a, bb, acc0);
      bb.x = p0[16]; bb.y = p1[16]; acc1 = wmma4(a, bb, acc1);
      bb.x = p0[32]; bb.y = p1[32]; acc2 = wmma4(a, bb, acc2);
      bb.x = p0[48]; bb.y = p1[48]; acc3 = wmma4(a, bb, acc3);
    }
  };

  stage(0, 0);
  wait_async();
  __syncthreads();
  int cur = 0;
  for (int kc = 0; kc < FEAT; kc += KC) {
    if (kc + KC < FEAT) stage(cur ^ 1, kc + KC);  // prefetch next chunk (async)
    compute(cur);
    wait_async();
    __syncthreads();
    cur ^= 1;
  }

  const int nl = lane & 15;
  const int rh = (lane >> 4) * 8;
  const int nBase = nBase0 + wave * 64;
  const float bias0 = bg[nBase + 0 * 16 + nl];
  const float bias1 = bg[nBase + 1 * 16 + nl];
  const float bias2 = bg[nBase + 2 * 16 + nl];
  const float bias3 = bg[nBase + 3 * 16 + nl];
#pragma unroll
  for (int v = 0; v < 8; ++v) {
    float* trow = T + (size_t)(mBase + rh + v) * ATTN + nBase;
    trow[0 * 16 + nl] = fmaxf(acc0[v] + bias0, 0.0f);
    trow[1 * 16 + nl] = fmaxf(acc1[v] + bias1, 0.0f);
    trow[2 * 16 + nl] = fmaxf(acc2[v] + bias2, 0.0f);
    trow[3 * 16 + nl] = fmaxf(acc3[v] + bias3, 0.0f);
  }
}

// ---------------------------------------------------------------------------
// Kernel C: row sums of the unmaterialized weight matrix -> inv[i].
// ---------------------------------------------------------------------------
__global__ void rowsum_kernel(const float* __restrict__ xywh,
                              const float* __restrict__ hx,
                              const float* __restrict__ hy,
                              const float* __restrict__ validf,
                              float* __restrict__ inv) {
  __shared__ float red[256];
  const int i = blockIdx.x;
  const int t = threadIdx.x;
  const float hxi = hx[i];
  const float hyi = hy[i];
  const float vi = validf[i];
  float s = 0.0f;
  for (int j = t; j < S_DIM; j += 256) {
    const float xj = xywh[j * 4 + 0];
    const float yj = xywh[j * 4 + 1];
    const float dx = hxi - xj;
    const float dy = hyi - yj;
    const float d = sqrtf(dx * dx + dy * dy);
    s += validf[j] / (d + EPS_DIS);
  }
  red[t] = s * vi;
  __syncthreads();
  for (int off = 128; off > 0; off >>= 1) {
    if (t < off) red[t] += red[t + off];
    __syncthreads();
  }
  if (t == 0) inv[i] = (red[0] > 0.0f) ? (1.0f / red[0]) : 0.0f;
}

// ---------------------------------------------------------------------------
// Kernel D: out = rownorm(W) @ targets (34 GFLOP, dominant).
// Block = (32,4): all 4 waves share one 64-col slice of `targets`, staged in
// LDS via async copies and reused 4x; each wave owns a different 16-row
// i-block. A-fragments (weights) are computed on the fly in VALU -> WMMA.
// ---------------------------------------------------------------------------
__global__ void out_gemm_kernel(const float* __restrict__ xywh,
                                const float* __restrict__ hx,
                                const float* __restrict__ hy,
                                const float* __restrict__ validf,
                                const float* __restrict__ inv,
                                const float* __restrict__ T,
                                float* __restrict__ out) {
  __shared__ float Tb[2][KJ * 64];    // KJ x 64

  const int lane = threadIdx.x;
  const int wave = threadIdx.y;
  const int tid = wave * 32 + lane;
  const int nBase = blockIdx.x * 64;
  const int iBase = blockIdx.y * 64 + wave * 16;
  const int il = lane & 15;
  const int kh = (lane >> 4) * 2;

  const int irow = iBase + il;
  const float hxi = hx[irow];
  const float hyi = hy[irow];
  const float vi = validf[irow];

  v8f acc0 = {}, acc1 = {}, acc2 = {}, acc3 = {};

  auto stage = [&](int b, int j0) {
    // KJ*64/4 float4 over 128 threads
#pragma unroll
    for (int t = 0; t < (KJ * 64 / 4) / 128; ++t) {
      const int q = t * 128 + tid;
      const int kr = q / 16;
      const int n4 = (q % 16) * 4;
      lds_copy_f4(T + (size_t)(j0 + kr) * ATTN + nBase + n4,
                  &Tb[b][kr * 64 + n4]);
    }
  };

  auto compute = [&](int b, int j0) {
#pragma unroll
    for (int k = 0; k < KJ; k += 4) {
      const int jj = j0 + k + kh;
      const float2 p0 = *(const float2*)(xywh + (size_t)jj * 4);
      const float2 p1 = *(const float2*)(xywh + (size_t)(jj + 1) * 4);
      const float dx0 = hxi - p0.x, dy0 = hyi - p0.y;
      const float dx1 = hxi - p1.x, dy1 = hyi - p1.y;
      const float w0 = vi * validf[jj] /
                       (sqrtf(dx0 * dx0 + dy0 * dy0) + EPS_DIS);
      const float w1 = vi * validf[jj + 1] /
                       (sqrtf(dx1 * dx1 + dy1 * dy1) + EPS_DIS);
      v2f a;
      a.x = w0;
      a.y = w1;
      const float* p = &Tb[b][0] + (k + kh) * 64 + il;
      const float* q = p + 64;
      v2f bb;
      bb.x = p[0];  bb.y = q[0];  acc0 = wmma4(a, bb, acc0);
      bb.x = p[16]; bb.y = q[16]; acc1 = wmma4(a, bb, acc1);
      bb.x = p[32]; bb.y = q[32]; acc2 = wmma4(a, bb, acc2);
      bb.x = p[48]; bb.y = q[48]; acc3 = wmma4(a, bb, acc3);
    }
  };

  stage(0, 0);
  wait_async();
  __syncthreads();
  int cur = 0;
  for (int j0 = 0; j0 < S_DIM; j0 += KJ) {
    if (j0 + KJ < S_DIM) stage(cur ^ 1, j0 + KJ);
    compute(cur, j0);
    wait_async();
    __syncthreads();
    cur ^= 1;
  }

  const int nl = lane & 15;
  const int rh = (lane >> 4) * 8;
#pragma unroll
  for (int v = 0; v < 8; ++v) {
    const int row = iBase + rh + v;
    const float sc = inv[row];
    float* orow = out + (size_t)row * ATTN + nBase;
    orow[0 * 16 + nl] = acc0[v] * sc;
    orow[1 * 16 + nl] = acc1[v] * sc;
    orow[2 * 16 + nl] = acc2[v] * sc;
    orow[3 * 16 + nl] = acc3[v] * sc;
  }
}

// ---------------------------------------------------------------------------
// Launch.
// Inputs: 0 batch_data 1 xywh 2 OW 3 OH 4 actor_weights 5 avg_pos
//         6 num_person 7 Wh 8 bh 9 Wg 10 bg
// d_out: targets_weighted [S,ATTN] then highlight_pred [S,2], flat f32.
// ---------------------------------------------------------------------------
extern "C" void kernel_launch(void* const* d_in, const int* in_sizes, int n_in,
                              void* d_out, int out_size, void* d_ws, size_t ws_size,
                              hipStream_t stream) {
  (void)in_sizes; (void)n_in; (void)out_size; (void)ws_size;

  const float* X = (const float*)d_in[0];
  const float* xywh = (const float*)d_in[1];
  const int* num_person = (const int*)d_in[6];
  const float* Wh = (const float*)d_in[7];
  const float* bh = (const float*)d_in[8];
  const float* Wg = (const float*)d_in[9];
  const float* bg = (const float*)d_in[10];

  float* out = (float*)d_out;
  float* out_hp = out + (size_t)S_DIM * ATTN;

  float* T = (float*)d_ws;                 // targets [S, ATTN]
  float* hx = T + (size_t)S_DIM * ATTN;
  float* hy = hx + S_DIM;
  float* validf = hy + S_DIM;
  float* inv = validf + S_DIM;

  highlight_kernel<<<S_DIM, 256, 0, stream>>>(X, xywh, Wh, bh, num_person,
                                              out_hp, hx, hy, validf);

  dim3 blk(32, 4);
  dim3 gridT(ATTN / 256, S_DIM / 16);
  targets_gemm_kernel<<<gridT, blk, 0, stream>>>(X, Wg, bg, T);

  rowsum_kernel<<<S_DIM, 256, 0, stream>>>(xywh, hx, hy, validf, inv);

  dim3 gridO(ATTN / 64, S_DIM / 64);
  out_gemm_kernel<<<gridO, blk, 0, stream>>>(xywh, hx, hy, validf, inv, T, out);
}